// ASGLoss_18872086298703
// MI455X (gfx1250) — compile-verified
//
#include <hip/hip_runtime.h>
#include <hip/hip_bf16.h>

#define BB 128
#define TT 1024
#define NN 128
#define SS 256
#define NEGV (-1e30f)
#define AS 132   // padded LDS row stride (128+4): conflict-free column walks

typedef __attribute__((ext_vector_type(2))) float v2f;
typedef __attribute__((ext_vector_type(8))) float v8f;

// ---------------- prep: E[i,j] = exp(transition[j,i]) (K-major for WMMA B) ----------
__global__ __launch_bounds__(256) void asg_prep_expT(const float* __restrict__ trans,
                                                     float* __restrict__ ET) {
    int idx = blockIdx.x * 256 + threadIdx.x;
    if (idx < NN * NN) {
        int i = idx >> 7, j = idx & 127;
        ET[i * NN + j] = __expf(trans[j * NN + i]);
    }
}

// ---------------- FCC: forward scan with f32 WMMA logsumexp-GEMM --------------------
__global__ __launch_bounds__(256) void asg_fcc(const float* __restrict__ x,
                                               const float* __restrict__ ET,
                                               const int* __restrict__ ilen,
                                               float* __restrict__ fcc) {
    __shared__ float alpha[16 * AS];
    __shared__ float pbuf[16 * AS];
    __shared__ float Mrow[16];

    const int tid   = threadIdx.x;
    const int b0    = blockIdx.x * 16;
    const int wave  = tid >> 5;
    const int lane  = tid & 31;
    const int hi    = lane >> 4;       // half-wave: 0 or 1
    const int row16 = lane & 15;
    const int jcol  = wave * 16 + row16;   // output column owned by this lane
    const int r     = tid >> 4;        // 0..15 row for reduction phases
    const int c     = tid & 15;

    // alpha0 = emissions at t=0
    for (int idx = tid; idx < 16 * NN; idx += 256) {
        int rr = idx >> 7, jj = idx & 127;
        alpha[rr * AS + jj] = x[(long)(b0 + rr) * TT * NN + jj];
    }

    // per-lane input lengths for the 8 accumulator rows (rv = v + 8*hi)
    int lenv[8];
#pragma unroll
    for (int v = 0; v < 8; ++v) lenv[v] = ilen[b0 + v + 8 * hi];

    // Preload this wave's B-matrix fragments (constant across all T steps) into VGPRs.
    // B layout (4x16 f32): VGPR v, lane L -> K = v + 2*(L/16), N = L%16.
    v2f bfrag[32];
#pragma unroll
    for (int kc = 0; kc < 32; ++kc) {
        int k0 = kc * 4 + 2 * hi;
        v2f t2;
        t2.x = ET[(k0 + 0) * NN + jcol];
        t2.y = ET[(k0 + 1) * NN + jcol];
        bfrag[kc] = t2;
    }
    __syncthreads();

    for (int t = 1; t < TT; ++t) {
        // prefetch emissions used at end of this step (overlaps with reduce+WMMA)
        float xv[8];
#pragma unroll
        for (int v = 0; v < 8; ++v)
            xv[v] = x[(long)(b0 + v + 8 * hi) * TT * NN + (long)t * NN + jcol];

        // phase 1: row max of alpha (16 rows x 16 threads, 8 cols each)
        float m = -3.0e38f;
#pragma unroll
        for (int k = 0; k < 8; ++k) m = fmaxf(m, alpha[r * AS + c + 16 * k]);
#pragma unroll
        for (int off = 8; off >= 1; off >>= 1) m = fmaxf(m, __shfl_xor(m, off, 16));
        if (c == 0) Mrow[r] = m;

        // phase 2: p = exp(alpha - rowmax)
#pragma unroll
        for (int k = 0; k < 8; ++k)
            pbuf[r * AS + c + 16 * k] = __expf(alpha[r * AS + c + 16 * k] - m);

        __syncthreads();

        // phase 3: (16x128) x (128x16) f32 GEMM via V_WMMA_F32_16X16X4_F32,
        // two independent accumulator chains to halve the RAW chain length.
        // A layout (16x4 f32): VGPR v, lane L -> M = L%16, K = v + 2*(L/16).
        v8f acc0 = {};
        v8f acc1 = {};
#pragma unroll
        for (int kc = 0; kc < 32; kc += 2) {
            v2f a0 = *(const v2f*)&pbuf[row16 * AS + kc * 4 + 2 * hi];
            acc0 = __builtin_amdgcn_wmma_f32_16x16x4_f32(
                false, a0, false, bfrag[kc], (short)0, acc0, false, false);
            v2f a1 = *(const v2f*)&pbuf[row16 * AS + (kc + 1) * 4 + 2 * hi];
            acc1 = __builtin_amdgcn_wmma_f32_16x16x4_f32(
                false, a1, false, bfrag[kc + 1], (short)0, acc1, false, false);
        }

        // phase 4: alpha_new = x[t] + M + log(sum), masked by t < input_length
        // D layout: VGPR v, lane L -> row = v + 8*(L/16), col = jcol
#pragma unroll
        for (int v = 0; v < 8; ++v) {
            int   rv  = v + 8 * hi;
            float old = alpha[rv * AS + jcol];
            float nv  = xv[v] + Mrow[rv] + __logf(acc0[v] + acc1[v]);
            alpha[rv * AS + jcol] = (t < lenv[v]) ? nv : old;
        }
        __syncthreads();
    }

    // fcc[b] = logsumexp_j alphaT[b, j]
    float m = -3.0e38f;
#pragma unroll
    for (int k = 0; k < 8; ++k) m = fmaxf(m, alpha[r * AS + c + 16 * k]);
#pragma unroll
    for (int off = 8; off >= 1; off >>= 1) m = fmaxf(m, __shfl_xor(m, off, 16));
    float s = 0.0f;
#pragma unroll
    for (int k = 0; k < 8; ++k) s += __expf(alpha[r * AS + c + 16 * k] - m);
#pragma unroll
    for (int off = 8; off >= 1; off >>= 1) s += __shfl_xor(s, off, 16);
    if (c == 0) fcc[b0 + r] = m + __logf(s);
}

// ---------------- FAC: forced-alignment banded scan --------------------------------
__device__ __forceinline__ float logaddexpf(float a, float b) {
    float m = fmaxf(a, b);
    return m + __logf(__expf(a - m) + __expf(b - m));
}

__global__ __launch_bounds__(256) void asg_fac(const float* __restrict__ x,
                                               const float* __restrict__ trans,
                                               const int* __restrict__ tgts,
                                               const int* __restrict__ ilen,
                                               const int* __restrict__ tlen,
                                               float* __restrict__ fac) {
    __shared__ float xrow[NN];
    __shared__ float bl[SS];

    const int b = blockIdx.x;
    const int s = threadIdx.x;

    const int   tgt     = tgts[b * SS + s];
    const int   prevl   = (s > 0) ? tgts[b * SS + s - 1] : tgt;
    const float self_tr = trans[tgt * NN + tgt];
    const float next_tr = trans[tgt * NN + prevl];
    const int   len     = ilen[b];

    if (s < NN) xrow[s] = x[(long)b * TT * NN + s];
    __syncthreads();
    float breg = (s == 0) ? xrow[tgt] : NEGV;
    bl[s] = breg;

    for (int t = 1; t < TT; ++t) {
        __syncthreads();                       // prev-iter bl writes & xrow reads done
        float bm1 = (s > 0) ? bl[s - 1] : NEGV;
        if (s < NN) xrow[s] = x[(long)b * TT * NN + (long)t * NN + s];
        __syncthreads();                       // xrow populated
        float em = xrow[tgt];
        float nv = em + logaddexpf(breg + self_tr, bm1 + next_tr);
        breg = (t < len) ? nv : breg;
        bl[s] = breg;
    }

    if (s == tlen[b] - 1) fac[b] = breg;
}

// ---------------- finalize: mean(fcc - fac) -----------------------------------------
__global__ __launch_bounds__(128) void asg_finalize(const float* __restrict__ fcc,
                                                    const float* __restrict__ fac,
                                                    float* __restrict__ out) {
    __shared__ float part[4];
    int   tid = threadIdx.x;
    float v   = fcc[tid] - fac[tid];
#pragma unroll
    for (int off = 16; off >= 1; off >>= 1) v += __shfl_xor(v, off, 32);
    if ((tid & 31) == 0) part[tid >> 5] = v;
    __syncthreads();
    if (tid == 0) out[0] = (part[0] + part[1] + part[2] + part[3]) * (1.0f / (float)BB);
}

// ---------------- launch -------------------------------------------------------------
extern "C" void kernel_launch(void* const* d_in, const int* in_sizes, int n_in,
                              void* d_out, int out_size, void* d_ws, size_t ws_size,
                              hipStream_t stream) {
    const float* trans = (const float*)d_in[0];
    const float* xin   = (const float*)d_in[1];
    const int*   tgts  = (const int*)d_in[2];
    const int*   ilen  = (const int*)d_in[3];
    const int*   tlen  = (const int*)d_in[4];
    float*       out   = (float*)d_out;

    float* ws  = (float*)d_ws;
    float* ET  = ws;                 // N*N f32
    float* fcc = ws + NN * NN;       // B f32
    float* fac = fcc + BB;           // B f32

    asg_prep_expT<<<(NN * NN + 255) / 256, 256, 0, stream>>>(trans, ET);
    asg_fcc<<<BB / 16, 256, 0, stream>>>(xin, ET, ilen, fcc);
    asg_fac<<<BB, 256, 0, stream>>>(xin, trans, tgts, ilen, tlen, fac);
    asg_finalize<<<1, 128, 0, stream>>>(fcc, fac, out);
}